// ModuleRenderScatter_910533067249
// MI455X (gfx1250) — compile-verified
//
#include <hip/hip_runtime.h>
#include <hip/hip_bf16.h>
#include <stdint.h>

#define TILE 16
#define RAD  5
#define HALO (TILE + 2 * RAD)        // 26
#define NPIX (HALO * HALO)           // 676

#define G8L2E 11.541560327111707f    // 8 * log2(e)  (2*GAMMA*log2e, GAMMA=4)
#define PI_F  3.14159265358979f

#if defined(__has_builtin)
#  if __has_builtin(__builtin_amdgcn_global_load_async_to_lds_b32)
#    define USE_ASYNC_LDS 1
#  endif
#  if __has_builtin(__builtin_amdgcn_s_wait_asynccnt)
#    define HAVE_WAIT_ASYNC_BUILTIN 1
#  endif
#endif

// Generic -> addrspace(1)/(3) int* conversion via integer round-trip.
// On AMDGPU a generic pointer into LDS carries the LDS byte offset in its
// low 32 bits, so the truncating cast yields a valid as(3) pointer.
__device__ __forceinline__ __attribute__((address_space(1))) int* as1(const void* p) {
  return (__attribute__((address_space(1))) int*)(uintptr_t)p;
}
__device__ __forceinline__ __attribute__((address_space(3))) int* as3(const void* p) {
  return (__attribute__((address_space(3))) int*)(uint32_t)(uintptr_t)p;
}

extern "C" __global__ void __launch_bounds__(256)
bokeh_render_kernel(const float* __restrict__ img,   // [B,3,H,W]
                    const float* __restrict__ dfc,   // [B,1,H,W]
                    float* __restrict__ out,         // [B,3,H,W]
                    int H, int W) {
  // Interleaved halo tile: (c0, c1, c2, E) with E = exp(-8*|defocus|)
  __shared__ float4 sPix[NPIX];
  // inv = 1/max(pi*r^2,1), or 0 when the source pixel is out of bounds
  __shared__ float  sInv[NPIX];

  const int tid = threadIdx.x;
  const int tx0 = blockIdx.x * TILE;
  const int ty0 = blockIdx.y * TILE;
  const int b   = blockIdx.z;
  const int gy0 = ty0 - RAD;
  const int gx0 = tx0 - RAD;

  const size_t plane = (size_t)H * (size_t)W;
  const float* imgb = img + (size_t)b * 3u * plane;
  const float* dfcb = dfc + (size_t)b * plane;

  // ---- Stage halo (clamped coordinates) into LDS via async global->LDS ----
  for (int i = tid; i < NPIX; i += 256) {
    const int hy = i / HALO;
    const int hx = i - hy * HALO;
    const int cy = min(max(gy0 + hy, 0), H - 1);
    const int cx = min(max(gx0 + hx, 0), W - 1);
    const size_t g = (size_t)cy * (size_t)W + (size_t)cx;
#if defined(USE_ASYNC_LDS)
    __builtin_amdgcn_global_load_async_to_lds_b32(as1(imgb + g),              as3(&sPix[i].x), 0, 0);
    __builtin_amdgcn_global_load_async_to_lds_b32(as1(imgb + plane + g),      as3(&sPix[i].y), 0, 0);
    __builtin_amdgcn_global_load_async_to_lds_b32(as1(imgb + 2u * plane + g), as3(&sPix[i].z), 0, 0);
    __builtin_amdgcn_global_load_async_to_lds_b32(as1(dfcb + g),              as3(&sPix[i].w), 0, 0);
#else
    sPix[i] = make_float4(imgb[g], imgb[plane + g], imgb[2u * plane + g], dfcb[g]);
#endif
  }
#if defined(USE_ASYNC_LDS)
#  if defined(HAVE_WAIT_ASYNC_BUILTIN)
  __builtin_amdgcn_s_wait_asynccnt(0);
#  else
  asm volatile("s_wait_asynccnt 0" ::: "memory");
#  endif
#endif
  __syncthreads();

  // ---- Derive per-source-pixel quantities in LDS ----
  // E = exp(-8r) carries both the sigmoid factor and (monotonically) r itself,
  // so the window test  max(|dy|,|dx|) <= floor(r)+1  <=>  r >= m-1  becomes
  // E <= exp(-8(m-1)) against a literal. No separate ir needed.
  for (int i = tid; i < NPIX; i += 256) {
    const int hy = i / HALO;
    const int hx = i - hy * HALO;
    const int gy = gy0 + hy;
    const int gx = gx0 + hx;
    const bool valid = (gy >= 0) && (gy < H) && (gx >= 0) && (gx < W);
    const float r = fabsf(sPix[i].w);
    sPix[i].w = __builtin_amdgcn_exp2f(-G8L2E * r);             // E = exp(-8r)
    sInv[i]   = valid ? (1.0f / fmaxf(PI_F * r * r, 1.0f)) : 0.0f;
  }
  __syncthreads();

  // ---- 121-tap gather per output pixel ----
  const int oy = tid >> 4;
  const int ox = tid & 15;

  float a0 = 0.0f, a1 = 0.0f, a2 = 0.0f, ws = 0.0f;

#pragma unroll
  for (int dy = -RAD; dy <= RAD; ++dy) {
    const int rowb = (oy + RAD + dy) * HALO + ox + RAD;
#pragma unroll
    for (int dx = -RAD; dx <= RAD; ++dx) {
      const int ady = dy < 0 ? -dy : dy;
      const int adx = dx < 0 ? -dx : dx;
      const int m   = ady > adx ? ady : adx;
      // Kd = exp(8*dist): literal after unroll (LLVM folds sqrt/exp of constants)
      const float Kd = __builtin_expf(8.0f * __builtin_sqrtf((float)(dy * dy + dx * dx)));
      const float4 p  = sPix[rowb + dx];
      const float inv = sInv[rowb + dx];
      // sigmoid(8(r-d)) = 1 / (1 + exp(8d)*exp(-8r)) = 1 / (1 + Kd*E)
      const float e  = Kd * p.w;
      const float rc = __builtin_amdgcn_rcpf(1.0f + e);
      float sel = inv;
      if (m > 1) {  // m<=1 always inside window (ir = floor(r)+1 >= 1)
        const float Km = __builtin_expf(-8.0f * (float)(m - 1));  // literal
        sel = (p.w <= Km) ? inv : 0.0f;
      }
      const float w = sel * rc;
      a0 = fmaf(w, p.x, a0);
      a1 = fmaf(w, p.y, a1);
      a2 = fmaf(w, p.z, a2);
      ws += w;
    }
  }

  const float iw = 1.0f / ws;
  const int oyg = ty0 + oy;
  const int oxg = tx0 + ox;
  float* ob = out + (size_t)b * 3u * plane + (size_t)oyg * (size_t)W + (size_t)oxg;
  ob[0]          = a0 * iw;
  ob[plane]      = a1 * iw;
  ob[2u * plane] = a2 * iw;
}

extern "C" void kernel_launch(void* const* d_in, const int* in_sizes, int n_in,
                              void* d_out, int out_size, void* d_ws, size_t ws_size,
                              hipStream_t stream) {
  const float* img = (const float*)d_in[0];   // [B,3,H,W] fp32
  const float* dfc = (const float*)d_in[1];   // [B,1,H,W] fp32
  float* out = (float*)d_out;                 // [B,3,H,W] fp32

  const int H = 512, W = 512, C = 3;
  const int B = in_sizes[0] / (C * H * W);

  dim3 grid(W / TILE, H / TILE, B);
  dim3 block(256, 1, 1);
  bokeh_render_kernel<<<grid, block, 0, stream>>>(img, dfc, out, H, W);
}